// KnowledgeGraphEmbeddingLayer_33483565040009
// MI455X (gfx1250) — compile-verified
//
#include <hip/hip_runtime.h>
#include <math.h>

typedef __attribute__((ext_vector_type(2))) float v2f;
typedef __attribute__((ext_vector_type(8))) float v8f;

constexpr int kN    = 65536;   // triplets
constexpr int kD    = 64;      // embedding dim
constexpr int kF    = 16;      // feature dim
constexpr int kK1   = 80;      // D + F (GEMM1 K-depth)
constexpr int kST   = 84;      // LDS row stride (floats) for 16x80 tiles (bank-conflict pad)
constexpr int kWAVES = 4;      // waves per block
constexpr int kTILE  = 16;     // triplets per wave

// per-wave LDS floats: CH(16*kST) + CT(16*kST) + P(16*17)
constexpr int kCH = 0;
constexpr int kCT = 16 * kST;
constexpr int kP  = 2 * 16 * kST;
constexpr int kWAVE_FLOATS = 2 * 16 * kST + 16 * 17;

__global__ __launch_bounds__(kWAVES * 32)
void kge_kernel(const float* __restrict__ x,
                const int*   __restrict__ trip,
                const float* __restrict__ ent,
                const float* __restrict__ rel_emb,
                const float* __restrict__ pmats,
                const float* __restrict__ W,
                const float* __restrict__ bvec,
                float* __restrict__ out_head,
                float* __restrict__ out_tail,
                float* __restrict__ out_r,
                float* __restrict__ out_sc)
{
    __shared__ float smem[kWAVES * kWAVE_FLOATS];
    __shared__ int   srel[kWAVES][16];
    __shared__ int   shid[kWAVES][16];
    __shared__ int   stid[kWAVES][16];

    const int w    = threadIdx.x >> 5;
    const int lane = threadIdx.x & 31;
    const int c    = lane & 15;   // column / M-row within half
    const int q    = lane >> 4;   // lane half

    float* CH = smem + w * kWAVE_FLOATS + kCH;   // [h | x] tile, later diff tile
    float* CT = smem + w * kWAVE_FLOATS + kCT;   // [t | x] tile
    float* P  = smem + w * kWAVE_FLOATS + kP;    // norm-reduction scratch

    const int tile = blockIdx.x * kWAVES + w;
    const int r0   = tile * kTILE;

    // ---- triplet metadata: rel + offset-adjusted head/tail entity rows ----
    if (lane < 16) {
        const int n  = r0 + lane;
        const int h0 = trip[3 * n + 0];
        const int t0 = trip[3 * n + 1];
        const int rl = trip[3 * n + 2];
        srel[w][lane] = rl;
        // head_off = {PLAYLIST(500000), TRACK(0), TRACK(0)}
        shid[w][lane] = (rl == 0 ? 500000 : 0) + h0;
        // tail_off = {TRACK(0), ARTIST(700000), ALBUM(600000)}
        stid[w][lane] = (rl == 0 ? 0 : (rl == 1 ? 700000 : 600000)) + t0;
    }

    // ---- stage [h|x] and [t|x] 16x80 tiles into LDS (row-major, stride kST) ----
    for (int i = 0; i < 16; ++i) {
        const long hrow = shid[w][i];
        const long trow = stid[w][i];
        v2f hv = *(const v2f*)(ent + hrow * kD + 2 * lane);
        v2f tv = *(const v2f*)(ent + trow * kD + 2 * lane);
        *(v2f*)&CH[i * kST + 2 * lane] = hv;
        *(v2f*)&CT[i * kST + 2 * lane] = tv;
        if (lane < 16) {
            const float xv = x[(r0 + i) * kF + lane];
            CH[i * kST + kD + lane] = xv;
            CT[i * kST + kD + lane] = xv;
        }
    }

    // ---- GEMM 1: proj = concat @ W^T  via V_WMMA_F32_16X16X4_F32 ----
    // A frag: row=c, k = k0 + 2*q + {0,1}.  B frag: B[k][j]=W[j][k], j = nt*16+c.
    v8f accH[4] = {};
    v8f accT[4] = {};
    #pragma unroll
    for (int s = 0; s < 20; ++s) {
        const int k0 = 4 * s;
        v2f ah = *(const v2f*)&CH[c * kST + k0 + 2 * q];
        v2f at = *(const v2f*)&CT[c * kST + k0 + 2 * q];
        #pragma unroll
        for (int nt = 0; nt < 4; ++nt) {
            v2f bw = *(const v2f*)(W + (nt * 16 + c) * kK1 + k0 + 2 * q);
            accH[nt] = __builtin_amdgcn_wmma_f32_16x16x4_f32(
                false, ah, false, bw, (short)0, accH[nt], false, false);
            accT[nt] = __builtin_amdgcn_wmma_f32_16x16x4_f32(
                false, at, false, bw, (short)0, accT[nt], false, false);
        }
    }

    int rv[8];
    #pragma unroll
    for (int v = 0; v < 8; ++v) rv[v] = srel[w][v + 8 * q];

    float bc[4];
    #pragma unroll
    for (int nt = 0; nt < 4; ++nt) bc[nt] = bvec[nt * 16 + c];

    // ---- select head_e/tail_e in C-layout, store, build diff tile in CH ----
    // C layout: row = v + 8*q, col = nt*16 + c
    #pragma unroll
    for (int nt = 0; nt < 4; ++nt) {
        #pragma unroll
        for (int v = 0; v < 8; ++v) {
            const int row = v + 8 * q;
            const int col = nt * 16 + c;
            const float hc = CH[row * kST + col];
            const float tc = CT[row * kST + col];
            const float ph = accH[nt][v] + bc[nt];
            const float pt = accT[nt][v] + bc[nt];
            const bool is0 = (rv[v] == 0);
            const float he = is0 ? hc : ph;
            const float te = is0 ? pt : tc;
            out_head[(size_t)(r0 + row) * kD + col] = he;
            out_tail[(size_t)(r0 + row) * kD + col] = te;
            CH[row * kST + col] = he - te;   // diff tile (reuse CH region)
        }
    }

    // ---- r output rows (gather relation_emb[rel]) ----
    for (int i = 0; i < 16; ++i) {
        const int rl = srel[w][i];
        v2f rvv = *(const v2f*)(rel_emb + rl * kD + 2 * lane);
        *(v2f*)(out_r + (size_t)(r0 + i) * kD + 2 * lane) = rvv;
    }

    // ---- GEMM 2: P_r = diff @ M_r^T for r=0..2, lane-select per row ----
    v8f accS[4] = {};
    for (int r = 0; r < 3; ++r) {
        const float* Mr = pmats + r * (kD * kD);
        v8f accP[4] = {};
        #pragma unroll
        for (int s = 0; s < 16; ++s) {
            const int k0 = 4 * s;
            v2f ad = *(const v2f*)&CH[c * kST + k0 + 2 * q];
            #pragma unroll
            for (int nt = 0; nt < 4; ++nt) {
                v2f bm = *(const v2f*)(Mr + (nt * 16 + c) * kD + k0 + 2 * q);
                accP[nt] = __builtin_amdgcn_wmma_f32_16x16x4_f32(
                    false, ad, false, bm, (short)0, accP[nt], false, false);
            }
        }
        #pragma unroll
        for (int nt = 0; nt < 4; ++nt)
            #pragma unroll
            for (int v = 0; v < 8; ++v)
                if (rv[v] == r) accS[nt][v] = accP[nt][v];
    }

    // ---- scores: ||M@diff + r||_2 per row ----
    #pragma unroll
    for (int v = 0; v < 8; ++v) {
        float acc = 0.f;
        #pragma unroll
        for (int nt = 0; nt < 4; ++nt) {
            const float re = rel_emb[rv[v] * kD + nt * 16 + c];
            const float sv = accS[nt][v] + re;
            acc += sv * sv;
        }
        P[(v + 8 * q) * 17 + c] = acc;
    }
    if (lane < 16) {
        float sum = 0.f;
        #pragma unroll
        for (int j = 0; j < 16; ++j) sum += P[lane * 17 + j];
        out_sc[r0 + lane] = sqrtf(sum);
    }
}

extern "C" void kernel_launch(void* const* d_in, const int* in_sizes, int n_in,
                              void* d_out, int out_size, void* d_ws, size_t ws_size,
                              hipStream_t stream) {
    (void)in_sizes; (void)n_in; (void)out_size; (void)d_ws; (void)ws_size;
    const float* x       = (const float*)d_in[0];
    const int*   trip    = (const int*)d_in[1];
    const float* ent     = (const float*)d_in[2];
    const float* rel_emb = (const float*)d_in[3];
    const float* pmats   = (const float*)d_in[4];
    const float* W       = (const float*)d_in[5];
    const float* bvec    = (const float*)d_in[6];

    float* out      = (float*)d_out;
    float* out_head = out;
    float* out_tail = out + (size_t)kN * kD;
    float* out_r    = out + (size_t)2 * kN * kD;
    float* out_sc   = out + (size_t)3 * kN * kD;

    dim3 grid(kN / (kWAVES * kTILE));   // 1024 blocks
    dim3 block(kWAVES * 32);            // 4 waves (wave32)
    hipLaunchKernelGGL(kge_kernel, grid, block, 0, stream,
                       x, trip, ent, rel_emb, pmats, W, bvec,
                       out_head, out_tail, out_r, out_sc);
}